// SpatialTransformer_2688649527742
// MI455X (gfx1250) — compile-verified
//
#include <hip/hip_runtime.h>

// Spatial transformer: batched 3D affine warp + trilinear interpolation.
// vol: [B,160,160,160,2] f32, trf: [B,3,4] f32, out: [B,160,160,160,2] f32.
//
// Memory-bound gather (AI ~ 2.8 FLOP/byte) -> optimize the MI455X memory path:
//  - whole input (131 MB) is L2-resident (192 MB L2) -> gathers hit L2
//  - output is streamed write-once -> non-temporal global_store_b64 so the
//    131 MB of stores don't evict the volume from L2
//  - float2 (8B) loads/stores -> global_load_b64 / global_store_b64
//  - global_prefetch_b8 of the identity-mapped line overlaps coordinate math
//  - 3D grid: k=threadIdx.x (coalesced, 5 wave32s), j/i/b from blockIdx ->
//    zero integer div/mod; i,j uniform -> affine row terms + row base
//    addresses fold onto the SALU, minimal per-lane VALU work.

typedef float v2f __attribute__((ext_vector_type(2)));

#define DIM_D 160
#define DIM_H 160
#define DIM_W 160
#define CENTER 79.5f                     // (160-1)/2
#define VOXELS (DIM_D * DIM_H * DIM_W)   // 4,096,000

__global__ __launch_bounds__(DIM_W) void spatial_transformer_kernel(
    const float* __restrict__ vol,   // [B,D,H,W,2]
    const float* __restrict__ trf,   // [B,3,4]
    float* __restrict__ out)         // [B,D,H,W,2]
{
    const int k = threadIdx.x;   // 0..159, lane-varying (coalesced along W)
    const int j = blockIdx.x;    // 0..159, uniform
    const int i = blockIdx.y;    // 0..159, uniform
    const int b = blockIdx.z;    // batch,  uniform

    const int row = (i * DIM_H + j) * DIM_W;           // uniform row base
    const float* __restrict__ volb = vol + (size_t)b * (VOXELS * 2);

    // Warm the identity-mapped cacheline (displacement is small near center)
    // while the scalar matrix loads + coordinate math are in flight.
    __builtin_prefetch(volb + (size_t)(row + k) * 2, 0, 3);

    // Affine matrix: pointer is grid-uniform -> s_load path.
    const float* __restrict__ m = trf + b * 12;
    const float m00 = m[0], m01 = m[1], m02 = m[2],  m03 = m[3];
    const float m10 = m[4], m11 = m[5], m12 = m[6],  m13 = m[7];
    const float m20 = m[8], m21 = m[9], m22 = m[10], m23 = m[11];

    // Sample location = M @ [centered coords, 1] + center  (reference math).
    // xc, yc are uniform; only zc varies per lane.
    const float xc = (float)i - CENTER;
    const float yc = (float)j - CENTER;
    const float zc = (float)k - CENTER;
    const float bx = fmaf(m00, xc, fmaf(m01, yc, m03)) + CENTER;  // uniform
    const float by = fmaf(m10, xc, fmaf(m11, yc, m13)) + CENTER;  // uniform
    const float bz = fmaf(m20, xc, fmaf(m21, yc, m23)) + CENTER;  // uniform
    float x = fmaf(m02, zc, bx);
    float y = fmaf(m12, zc, by);
    float z = fmaf(m22, zc, bz);

    // clip to [0, dim-1]
    x = fminf(fmaxf(x, 0.f), (float)(DIM_D - 1));
    y = fminf(fmaxf(y, 0.f), (float)(DIM_H - 1));
    z = fminf(fmaxf(z, 0.f), (float)(DIM_W - 1));

    const float x0f = floorf(x), y0f = floorf(y), z0f = floorf(z);
    const float x1f = fminf(x0f + 1.f, (float)(DIM_D - 1));
    const float y1f = fminf(y0f + 1.f, (float)(DIM_H - 1));
    const float z1f = fminf(z0f + 1.f, (float)(DIM_W - 1));

    // voxelmorph convention: w_lo = loc1 - loc, w_hi = 1 - w_lo
    const float wx0 = x1f - x, wy0 = y1f - y, wz0 = z1f - z;
    const float wx1 = 1.f - wx0, wy1 = 1.f - wy0, wz1 = 1.f - wz0;

    const int x0 = (int)x0f, y0 = (int)y0f, z0 = (int)z0f;
    const int x1 = (int)x1f, y1 = (int)y1f, z1 = (int)z1f;

    // 8 trilinear corners, each a single aligned global_load_b64 (2 channels)
    const v2f* __restrict__ v2 = (const v2f*)volb;
    const int r00 = (x0 * DIM_H + y0) * DIM_W;
    const int r01 = (x0 * DIM_H + y1) * DIM_W;
    const int r10 = (x1 * DIM_H + y0) * DIM_W;
    const int r11 = (x1 * DIM_H + y1) * DIM_W;

    const v2f c000 = v2[r00 + z0], c001 = v2[r00 + z1];
    const v2f c010 = v2[r01 + z0], c011 = v2[r01 + z1];
    const v2f c100 = v2[r10 + z0], c101 = v2[r10 + z1];
    const v2f c110 = v2[r11 + z0], c111 = v2[r11 + z1];

    const float w00 = wx0 * wy0, w01 = wx0 * wy1;
    const float w10 = wx1 * wy0, w11 = wx1 * wy1;
    const float w000 = w00 * wz0, w001 = w00 * wz1;
    const float w010 = w01 * wz0, w011 = w01 * wz1;
    const float w100 = w10 * wz0, w101 = w10 * wz1;
    const float w110 = w11 * wz0, w111 = w11 * wz1;

    v2f acc;
    acc.x = fmaf(c000.x, w000, fmaf(c001.x, w001,
            fmaf(c010.x, w010, fmaf(c011.x, w011,
            fmaf(c100.x, w100, fmaf(c101.x, w101,
            fmaf(c110.x, w110, c111.x * w111)))))));
    acc.y = fmaf(c000.y, w000, fmaf(c001.y, w001,
            fmaf(c010.y, w010, fmaf(c011.y, w011,
            fmaf(c100.y, w100, fmaf(c101.y, w101,
            fmaf(c110.y, w110, c111.y * w111)))))));

    // Streaming write-once output: non-temporal b64 store preserves L2
    // residency of the input volume. Address = SGPR base + k*8.
    v2f* __restrict__ o = (v2f*)out + (size_t)b * VOXELS + (row + k);
    __builtin_nontemporal_store(acc, o);
}

extern "C" void kernel_launch(void* const* d_in, const int* in_sizes, int n_in,
                              void* d_out, int out_size, void* d_ws, size_t ws_size,
                              hipStream_t stream) {
    const float* vol = (const float*)d_in[0];   // [B,160,160,160,2]
    const float* trf = (const float*)d_in[1];   // [B,3,4]
    float* out = (float*)d_out;

    const int B = in_sizes[1] / 12;             // 4

    dim3 grid(DIM_H, DIM_D, B);                 // j, i, b
    spatial_transformer_kernel<<<grid, DIM_W, 0, stream>>>(vol, trf, out);
}